// GCNBlock_20744692039823
// MI455X (gfx1250) — compile-verified
//
#include <hip/hip_runtime.h>
#include <hip/hip_bf16.h>

typedef __attribute__((ext_vector_type(2))) float v2f;
typedef __attribute__((ext_vector_type(8))) float v8f;

#define DIM 128

// ---------------------------------------------------------------- utilities
__global__ void fill_f32(float* p, float v, int n) {
    int i = blockIdx.x * blockDim.x + threadIdx.x;
    if (i < n) p[i] = v;
}

// deg starts at 1.0 (self loop); add incoming edge weights at col
__global__ void deg_edges(const int* __restrict__ col, const float* __restrict__ w,
                          float* __restrict__ deg, int nE) {
    int e = blockIdx.x * blockDim.x + threadIdx.x;
    if (e < nE) unsafeAtomicAdd(&deg[col[e]], w[e]);
}

__global__ void deg_to_dinv(float* __restrict__ d, int n) {
    int i = blockIdx.x * blockDim.x + threadIdx.x;
    if (i < n) {
        float v = d[i];
        d[i] = v > 0.f ? __frsqrt_rn(v) : 0.f;
    }
}

__global__ void count_nodes(const int* __restrict__ batch, float* __restrict__ counts, int n) {
    int i = blockIdx.x * blockDim.x + threadIdx.x;
    if (i < n) unsafeAtomicAdd(&counts[batch[i]], 1.0f);
}

// ------------------------------------------------------- WMMA fp32 GEMM
// O[nRows x 128] = H[nRows x 128] @ W[128 x 128]
// One wave32 computes one 16x16 tile; K=128 swept as 32 x (K=4) WMMA steps.
// A 16x4 f32 fragment (ISA 7.12.2): lanes 0-15 hold K=0,1; lanes 16-31 hold K=2,3
// (row M = lane&15 within a lane's 2 VGPRs). B 4x16 mirrors it across lanes.
__global__ __launch_bounds__(256)
void gemm_wmma_f32(const float* __restrict__ H, const float* __restrict__ W,
                   float* __restrict__ O, int nRows) {
    const int lane  = threadIdx.x & 31;
    const int wave  = threadIdx.x >> 5;
    const int tileM = blockIdx.x;         // 8 waves = 8 N-tiles of one M-tile
    const int tileN = wave;
    const int mbase = tileM * 16;
    if (mbase >= nRows) return;
    const int nbase = tileN * 16;
    const int mrow  = lane & 15;
    const int koff  = (lane >> 4) << 1;   // 0 or 2

    v8f acc = {};
    const float* arow = H + (size_t)(mbase + mrow) * DIM + koff;
    const float* bcol = W + (size_t)koff * DIM + nbase + (lane & 15);
#pragma unroll
    for (int k = 0; k < DIM; k += 4) {
        v2f a = *(const v2f*)(arow + k);          // A[m][k+koff], A[m][k+koff+1]
        v2f b;
        b.x = bcol[(size_t)k * DIM];              // B[k+koff][n]
        b.y = bcol[(size_t)k * DIM + DIM];        // B[k+koff+1][n]
        acc = __builtin_amdgcn_wmma_f32_16x16x4_f32(
            /*neg_a=*/false, a, /*neg_b=*/false, b,
            /*c_mod=*/(short)0, acc, /*reuse_a=*/false, /*reuse_b=*/false);
    }
    // C/D layout: VGPR r -> M = r (lanes 0-15) / r+8 (lanes 16-31); N = lane&15
    const int radd = (lane >> 4) << 3;
    float* orow = O + (size_t)(mbase + radd) * DIM + nbase + (lane & 15);
#pragma unroll
    for (int r = 0; r < 8; ++r) orow[(size_t)r * DIM] = acc[r];
}

// ----------------------------------------------- message passing (scatter)
// agg[i][:] = hl[i][:] * dinv[i]^2   (self-loop term; fully overwrites agg)
__global__ void self_init(const float* __restrict__ hl, const float* __restrict__ dinv,
                          float* __restrict__ agg, int total) {
    int i = blockIdx.x * blockDim.x + threadIdx.x;
    if (i < total) {
        float di = dinv[i >> 7];
        agg[i] = hl[i] * di * di;
    }
}

// one wave32 per edge; 4 floats/lane covers DIM=128
__global__ __launch_bounds__(256)
void edge_scatter(const int* __restrict__ row, const int* __restrict__ col,
                  const float* __restrict__ w, const float* __restrict__ dinv,
                  const float* __restrict__ hl, float* __restrict__ agg, int nE) {
    int e = blockIdx.x * 8 + (threadIdx.x >> 5);
    if (e >= nE) return;
    int lane = threadIdx.x & 31;
    int r = row[e], c = col[e];
    float nrm = dinv[r] * w[e] * dinv[c];
    const float4 v = *(const float4*)(hl + (size_t)r * DIM + lane * 4);
    float* dst = agg + (size_t)c * DIM + lane * 4;
    unsafeAtomicAdd(dst + 0, v.x * nrm);
    unsafeAtomicAdd(dst + 1, v.y * nrm);
    unsafeAtomicAdd(dst + 2, v.z * nrm);
    unsafeAtomicAdd(dst + 3, v.w * nrm);
}

// h = relu(agg + b) in place; pool-accumulate into emb[batch[n]]
__global__ void bias_relu_pool(float* __restrict__ agg, const float* __restrict__ b,
                               const int* __restrict__ batch, float* __restrict__ emb,
                               int total) {
    int i = blockIdx.x * blockDim.x + threadIdx.x;
    if (i < total) {
        int d = i & (DIM - 1);
        int n = i >> 7;
        float v = fmaxf(agg[i] + b[d], 0.0f);
        agg[i] = v;
        unsafeAtomicAdd(&emb[(size_t)batch[n] * DIM + d], v);
    }
}

__global__ void finalize_mean(float* __restrict__ out, const float* __restrict__ counts,
                              int total) {
    int i = blockIdx.x * blockDim.x + threadIdx.x;
    if (i < total) {
        int g = (i >> 7) & 255;
        out[i] /= fmaxf(counts[g], 1.0f);
    }
}

// ------------------------------------------------------------------ launch
extern "C" void kernel_launch(void* const* d_in, const int* in_sizes, int n_in,
                              void* d_out, int out_size, void* d_ws, size_t ws_size,
                              hipStream_t stream) {
    const float* x     = (const float*)d_in[0];
    const int*   eidx  = (const int*)d_in[1];
    const float* ew    = (const float*)d_in[2];
    const int*   batch = (const int*)d_in[3];
    const float* Wm[3] = {(const float*)d_in[4], (const float*)d_in[6], (const float*)d_in[8]};
    const float* bv[3] = {(const float*)d_in[5], (const float*)d_in[7], (const float*)d_in[9]};
    float* out = (float*)d_out;

    const int N = in_sizes[0] / DIM;   // 100000
    const int E = in_sizes[2];         // 1600000
    const int GD = out_size / 3;       // 256*128 per layer embedding
    const int ND = N * DIM;

    const int* erow = eidx;
    const int* ecol = eidx + E;

    // workspace layout (256B aligned)
    auto align256 = [](size_t v) { return (v + 255) & ~(size_t)255; };
    char* ws = (char*)d_ws;
    float* dinv   = (float*)ws;                                   size_t off = align256((size_t)N * 4);
    float* counts = (float*)(ws + off);                           off += align256(256 * 4);
    float* hl     = (float*)(ws + off);                           off += align256((size_t)ND * 4);
    float* hbuf   = (float*)(ws + off);                           // ping buffer / scatter dest

    const int T = 256;
    dim3 blk(T);

    // degrees (start at 1.0 = self loop), counts, output accumulators
    fill_f32<<<dim3((N + T - 1) / T), blk, 0, stream>>>(dinv, 1.0f, N);
    fill_f32<<<dim3(1), blk, 0, stream>>>(counts, 0.0f, 256);
    fill_f32<<<dim3((out_size + T - 1) / T), blk, 0, stream>>>(out, 0.0f, out_size);
    deg_edges<<<dim3((E + T - 1) / T), blk, 0, stream>>>(ecol, ew, dinv, E);
    deg_to_dinv<<<dim3((N + T - 1) / T), blk, 0, stream>>>(dinv, N);
    count_nodes<<<dim3((N + T - 1) / T), blk, 0, stream>>>(batch, counts, N);

    const int gemmBlocks = (N + 15) / 16;       // 8 waves/block -> 8 N-tiles each
    const int ndBlocks   = (ND + T - 1) / T;
    const int edgeBlocks = (E + 7) / 8;

    for (int l = 0; l < 3; ++l) {
        const float* hin = (l == 0) ? x : hbuf;
        // hl = hin @ W  (WMMA f32)
        gemm_wmma_f32<<<dim3(gemmBlocks), blk, 0, stream>>>(hin, Wm[l], hl, N);
        // agg (= hbuf, hin no longer needed): self loop then edge scatter
        self_init<<<dim3(ndBlocks), blk, 0, stream>>>(hl, dinv, hbuf, ND);
        edge_scatter<<<dim3(edgeBlocks), blk, 0, stream>>>(erow, ecol, ew, dinv, hl, hbuf, E);
        // h = relu(agg + b); pool into this layer's embedding slab
        bias_relu_pool<<<dim3(ndBlocks), blk, 0, stream>>>(hbuf, bv[l], batch,
                                                           out + (size_t)l * GD, ND);
    }
    finalize_mean<<<dim3((out_size + T - 1) / T), blk, 0, stream>>>(out, counts, out_size);
}